// ENNWithSimpleBICEP_22445499089476
// MI455X (gfx1250) — compile-verified
//
#include <hip/hip_runtime.h>
#include <math.h>

// ---------------------------------------------------------------------------
// Types for WMMA fragments (gfx1250, wave32)
// ---------------------------------------------------------------------------
typedef float v2f __attribute__((ext_vector_type(2)));
typedef float v8f __attribute__((ext_vector_type(8)));

// ---------------------------------------------------------------------------
// Kernel 1: noise_sum[b,p] = sqrt(dt) * sum_s noise[b,p,s]   (s = 50)
// One wave32 per row; lanes read contiguous floats -> coalesced.
// ---------------------------------------------------------------------------
__global__ __launch_bounds__(256) void noise_reduce(const float* __restrict__ noise,
                                                    float* __restrict__ ns,
                                                    int rows, float sqrt_dt) {
    int gw   = (int)((blockIdx.x * blockDim.x + threadIdx.x) >> 5);
    int lane = threadIdx.x & 31;
    if (gw >= rows) return;
    const float* p = noise + (size_t)gw * 50;
    float s = p[lane];                      // lanes 0..31 (< 50) always valid
    if (lane + 32 < 50) s += p[lane + 32];  // lanes 0..17
    #pragma unroll
    for (int off = 16; off > 0; off >>= 1) s += __shfl_down(s, off);
    if (lane == 0) ns[gw] = s * sqrt_dt;
}

// ---------------------------------------------------------------------------
// Kernel 2: f32 WMMA GEMM:  C = act( A[MxK] * B[KxN] + bias + add )
//   block tile : 128(M) x 128(N), 256 threads = 8 waves (4 in M x 2 in N)
//   wave tile  : 32(M) x 64(N) = 2 x 4 WMMA tiles, 8 x v8f accumulators
//   K staged through double-buffered LDS in steps of 16 (register staging),
//   inner WMMA K-step = 4  ->  v_wmma_f32_16x16x4_f32
// Requires: M % 128 == 0, N % 128 == 0, K % 16 == 0 (true for this problem).
//
// Fragment layouts (CDNA5 ISA 7.12.2, wave32):
//   A (16x4 f32):  lane L -> M = L%16 ; VGPR v -> K = 2*(L/16) + v
//   B (4x16 f32):  lane L -> N = L%16 ; VGPR v -> K = 2*(L/16) + v
//   C/D (16x16):   lane L -> N = L%16 ; VGPR v -> M = v + 8*(L/16)
//
// LDS layouts:
//   As[m][k]            stride LDA_S=20  -> b64 A-frag reads conflict-free
//   Bp[k/2][2n + k&1]   stride PSTR=288  -> single aligned b64 per B-frag,
//                        288%64==32 puts the upper lane-half on opposite banks
// ---------------------------------------------------------------------------
#define BM 128
#define BN 128
#define BK 16
#define LDA_S 20
#define PSTR (2 * BN + 32) /* 288 */

__device__ __forceinline__ void load_tile_regs(
    const float* __restrict__ A, const float* __restrict__ B,
    int N, int K, int bm, int bn, int k0, int tid,
    float4 ra[2], float4 rb[2])
{
    #pragma unroll
    for (int i = 0; i < 2; ++i) {
        int e  = tid + i * 256;
        int m  = e >> 2;                 // A: 4 float4 per 16-wide row
        int k4 = (e & 3) << 2;
        ra[i] = *reinterpret_cast<const float4*>(A + (size_t)(bm + m) * K + k0 + k4);
        int kb = e >> 5;                 // B: 32 float4 per 128-wide row
        int n4 = (e & 31) << 2;
        rb[i] = *reinterpret_cast<const float4*>(B + (size_t)(k0 + kb) * N + bn + n4);
    }
}

__device__ __forceinline__ void store_tile_lds(
    float* __restrict__ As, float* __restrict__ Bp, int tid,
    const float4 ra[2], const float4 rb[2])
{
    #pragma unroll
    for (int i = 0; i < 2; ++i) {
        int e  = tid + i * 256;
        int m  = e >> 2;
        int k4 = (e & 3) << 2;
        *reinterpret_cast<float4*>(&As[m * LDA_S + k4]) = ra[i];
        int kb = e >> 5;
        int n4 = (e & 31) << 2;
        const float* rv = reinterpret_cast<const float*>(&rb[i]);
        #pragma unroll
        for (int c = 0; c < 4; ++c)
            Bp[(kb >> 1) * PSTR + 2 * (n4 + c) + (kb & 1)] = rv[c];
    }
}

__device__ __forceinline__ void compute_block(
    const float* __restrict__ As, const float* __restrict__ Bp,
    int wave_m, int wave_n, int lane_m, int lane_k2, v8f acc[8])
{
    #pragma unroll
    for (int ks = 0; ks < BK; ks += 4) {
        v2f a[2];
        #pragma unroll
        for (int mt = 0; mt < 2; ++mt) {
            const float* ap =
                &As[(wave_m * 32 + mt * 16 + lane_m) * LDA_S + ks + lane_k2];
            a[mt][0] = ap[0];
            a[mt][1] = ap[1];
        }
        #pragma unroll
        for (int t = 0; t < 4; ++t) {
            const v2f b = *reinterpret_cast<const v2f*>(
                &Bp[((ks + lane_k2) >> 1) * PSTR +
                    2 * (wave_n * 64 + t * 16 + lane_m)]);
            acc[0 * 4 + t] = __builtin_amdgcn_wmma_f32_16x16x4_f32(
                false, a[0], false, b, (short)0, acc[0 * 4 + t], false, false);
            acc[1 * 4 + t] = __builtin_amdgcn_wmma_f32_16x16x4_f32(
                false, a[1], false, b, (short)0, acc[1 * 4 + t], false, false);
        }
    }
}

__global__ __launch_bounds__(256) void wmma_gemm_f32(
    const float* __restrict__ A, const float* __restrict__ B,
    const float* __restrict__ bias, const float* __restrict__ add,
    float* __restrict__ C, int M, int N, int K, int relu)
{
    __shared__ __align__(16) float As[2][BM * LDA_S];
    __shared__ __align__(16) float Bp[2][(BK / 2) * PSTR];

    const int tid     = threadIdx.x;
    const int wave    = tid >> 5;
    const int lane    = tid & 31;
    const int wave_m  = wave & 3;         // 4 waves in M
    const int wave_n  = wave >> 2;        // 2 waves in N
    const int lane_m  = lane & 15;        // M (A) / N (B,C) index
    const int lane_k2 = (lane >> 4) << 1; // K base from lane half
    const int lane_h8 = (lane >> 4) << 3; // M offset for C/D rows

    const int bm = blockIdx.y * BM;
    const int bn = blockIdx.x * BN;

    v8f acc[8];
    #pragma unroll
    for (int t = 0; t < 8; ++t)
        #pragma unroll
        for (int v = 0; v < 8; ++v) acc[t][v] = 0.0f;

    // ---- prologue: stage first K-block ----
    float4 ra[2], rb[2];
    load_tile_regs(A, B, N, K, bm, bn, 0, tid, ra, rb);
    store_tile_lds(&As[0][0], &Bp[0][0], tid, ra, rb);
    __syncthreads();

    // ---- double-buffered main loop ----
    int cur = 0;
    for (int k0 = BK; k0 < K; k0 += BK) {
        load_tile_regs(A, B, N, K, bm, bn, k0, tid, ra, rb);   // prefetch next
        compute_block(&As[cur][0], &Bp[cur][0],
                      wave_m, wave_n, lane_m, lane_k2, acc);   // overlap
        store_tile_lds(&As[cur ^ 1][0], &Bp[cur ^ 1][0], tid, ra, rb);
        __syncthreads();
        cur ^= 1;
    }
    compute_block(&As[cur][0], &Bp[cur][0],
                  wave_m, wave_n, lane_m, lane_k2, acc);

    // ---- epilogue: bias + optional residual add + optional ReLU ----
    #pragma unroll
    for (int mt = 0; mt < 2; ++mt) {
        #pragma unroll
        for (int t = 0; t < 4; ++t) {
            int n = bn + wave_n * 64 + t * 16 + lane_m;
            float bv = bias ? bias[n] : 0.0f;
            #pragma unroll
            for (int v = 0; v < 8; ++v) {
                int m = bm + wave_m * 32 + mt * 16 + lane_h8 + v;
                float val = acc[mt * 4 + t][v] + bv;
                if (add) val += add[(size_t)m * N + n];
                if (relu) val = val > 0.0f ? val : 0.0f;
                C[(size_t)m * N + n] = val;
            }
        }
    }
}

// ---------------------------------------------------------------------------
// Kernel 3: feedback + endpoints (fused), one wave32 per batch row
//   fb   = sigmoid(dot(h2[b,:], W_fb) + b_fb)
//   ep[b,p] = fb * drift_integral + ns[b,p]     (in-place on ns buffer)
// ---------------------------------------------------------------------------
__global__ __launch_bounds__(256) void feedback_endpoints(
    const float* __restrict__ h2, const float* __restrict__ W_fb,
    const float* __restrict__ b_fb, float* __restrict__ ep,
    int Bn, int H, int P, float drift)
{
    int gw   = (int)((blockIdx.x * blockDim.x + threadIdx.x) >> 5);
    int lane = threadIdx.x & 31;
    if (gw >= Bn) return;

    const float* hrow = h2 + (size_t)gw * H;
    float s = 0.0f;
    for (int k = lane; k < H; k += 32) s += hrow[k] * W_fb[k];
    #pragma unroll
    for (int off = 16; off > 0; off >>= 1) s += __shfl_down(s, off);
    s = __shfl(s, 0);
    float fb = 1.0f / (1.0f + expf(-(s + b_fb[0])));

    float* row = ep + (size_t)gw * P;
    for (int p = lane; p < P; p += 32) row[p] = fb * drift + row[p];
}

// ---------------------------------------------------------------------------
// Host-side launcher
// Inputs (all f32): x, W_in, b_in, W_hid, b_hid, W_fb, b_fb, W_agg, b_agg,
//                   W_out, b_out, noise
// ---------------------------------------------------------------------------
extern "C" void kernel_launch(void* const* d_in, const int* in_sizes, int n_in,
                              void* d_out, int out_size, void* d_ws, size_t ws_size,
                              hipStream_t stream) {
    const float* x     = (const float*)d_in[0];
    const float* W_in  = (const float*)d_in[1];
    const float* b_in  = (const float*)d_in[2];
    const float* W_hid = (const float*)d_in[3];
    const float* b_hid = (const float*)d_in[4];
    const float* W_fb  = (const float*)d_in[5];
    const float* b_fb  = (const float*)d_in[6];
    const float* W_agg = (const float*)d_in[7];
    const float* b_agg = (const float*)d_in[8];
    const float* W_out = (const float*)d_in[9];
    const float* b_out = (const float*)d_in[10];
    const float* noise = (const float*)d_in[11];

    const int Bsz = 8192, H = 1024, Din = 1024, Dout = 1024, P = 256, S = 50;

    float* h1 = (float*)d_ws;              // [B,H]   32 MB  (reused for "combined")
    float* h2 = h1 + (size_t)Bsz * H;      // [B,H]   32 MB
    float* ep = h2 + (size_t)Bsz * H;      // [B,P]    8 MB  (noise_sum, then endpoints)

    // Deterministic host-side scalar precompute (Euler drift integral)
    const float dt = 1.0f / (float)S;
    const float sqrt_dt = sqrtf(dt);
    double di = 0.0;
    for (int i = 0; i < S; ++i) di += exp(-0.1 * (double)i * (double)dt);
    const float drift = (float)(di * (double)dt);

    dim3 blk(256);

    // 1) noise reduction (independent of GEMM chain; bandwidth-dominant)
    {
        int rows = Bsz * P;                          // 2,097,152 rows
        int blocks = (rows + 7) / 8;                 // 8 waves per block
        noise_reduce<<<dim3(blocks), blk, 0, stream>>>(noise, ep, rows, sqrt_dt);
    }

    // 2) h1 = relu(x @ W_in + b_in)
    wmma_gemm_f32<<<dim3(H / BN, Bsz / BM), blk, 0, stream>>>(
        x, W_in, b_in, nullptr, h1, Bsz, H, Din, 1);

    // 3) h2 = relu(h1 @ W_hid + b_hid)
    wmma_gemm_f32<<<dim3(H / BN, Bsz / BM), blk, 0, stream>>>(
        h1, W_hid, b_hid, nullptr, h2, Bsz, H, H, 1);

    // 4) feedback + endpoints (in-place on ep)
    feedback_endpoints<<<dim3(Bsz / 8), blk, 0, stream>>>(
        h2, W_fb, b_fb, ep, Bsz, H, P, drift);

    // 5) combined = ep @ W_agg + b_agg + h2   (into h1 buffer, now dead)
    wmma_gemm_f32<<<dim3(H / BN, Bsz / BM), blk, 0, stream>>>(
        ep, W_agg, b_agg, h2, h1, Bsz, H, P, 0);

    // 6) out = combined @ W_out + b_out
    wmma_gemm_f32<<<dim3(Dout / BN, Bsz / BM), blk, 0, stream>>>(
        h1, W_out, b_out, nullptr, (float*)d_out, Bsz, Dout, H, 0);
}